// MultiHeadGraphAttention_24086176596039
// MI455X (gfx1250) — compile-verified
//
#include <hip/hip_runtime.h>
#include <hip/hip_bf16.h>
#include <math.h>

typedef __attribute__((ext_vector_type(2))) float v2f;
typedef __attribute__((ext_vector_type(8))) float v8f;

#define NEG_SLOPE 0.2f
#define LN_EPS 1e-5f

// ---------------------------------------------------------------------------
// LDS swizzle for the 128x128 transposed weight (Wt[col][d]).
// Plain stride-128 would put all 16 active lanes of a B-fragment load on the
// same bank (128 dwords == 0 mod 64 banks). Rotating each row by 2*col dwords
// spreads consecutive cols onto banks 2 apart. Because the B fragment's inner
// index d is always even, the rotated pair (d, d+1) never wraps -> each
// fragment is one aligned ds_load_b64.
// ---------------------------------------------------------------------------
__device__ __forceinline__ int swz(int c, int d) {
    return c * 128 + ((d + 2 * c) & 127);
}

// float atomic-max via sign-aware integer ordering (buffer init to -inf).
__device__ __forceinline__ void atomicMaxF(float* addr, float val) {
    if (val >= 0.0f) {
        atomicMax((int*)addr, __float_as_int(val));
    } else {
        atomicMin((unsigned int*)addr, (unsigned int)__float_as_int(val));
    }
}

// ---------------------------------------------------------------------------
// Kernel 1: xp[n, h*32+f] = sum_d x[n,d] * W[h,d,f]
// f32 WMMA 16x16x4. One wave computes 16 rows x 128 cols.
// ---------------------------------------------------------------------------
__global__ __launch_bounds__(256) void gat_gemm_xp(
    const float* __restrict__ x, const float* __restrict__ W,
    float* __restrict__ xp, int nrows)
{
    __shared__ float Wt[128 * 128];   // Wt[c][d] = W[c>>5][d][c&31], swizzled
    int tid = threadIdx.x;
    for (int i = tid; i < 128 * 128; i += 256) {
        int c = i >> 7, d = i & 127;
        Wt[swz(c, d)] = W[(((c >> 5) * 128) + d) * 32 + (c & 31)];
    }
    __syncthreads();

    int lane  = tid & 31;
    int wave  = tid >> 5;
    int rb    = (blockIdx.x * 8 + wave) * 16;
    if (rb >= nrows) return;          // wave-uniform exit, no barriers follow

    int m     = lane & 15;            // A row / B col within tile
    int khalf = lane >> 4;            // 0: K={0,1}, 1: K={2,3}

    v8f acc[8];
#pragma unroll
    for (int t = 0; t < 8; ++t) acc[t] = (v8f)0.0f;

    int rload = rb + m; if (rload >= nrows) rload = nrows - 1;
    const float* xr = x + (size_t)rload * 128 + 2 * khalf;

    for (int k = 0; k < 128; k += 4) {
        v2f a;
        a.x = xr[k];
        a.y = xr[k + 1];
        int d0 = k + 2 * khalf;       // even -> swizzled pair contiguous
#pragma unroll
        for (int t = 0; t < 8; ++t) {
            int col = t * 16 + m;
            v2f b = *(const v2f*)&Wt[swz(col, d0)];   // one ds_load_b64
            acc[t] = __builtin_amdgcn_wmma_f32_16x16x4_f32(
                false, a, false, b, (short)0, acc[t], false, false);
        }
    }

    // C/D layout: VGPR v -> row v (lanes 0-15) / row v+8 (lanes 16-31)
    int roff = rb + khalf * 8;
#pragma unroll
    for (int v = 0; v < 8; ++v) {
        int r = roff + v;
        if (r < nrows) {
#pragma unroll
            for (int t = 0; t < 8; ++t)
                xp[(size_t)r * 128 + t * 16 + m] = acc[t][v];
        }
    }
}

// ---------------------------------------------------------------------------
// Kernel 2: alpha_src/alpha_dst per (node, head)
// ---------------------------------------------------------------------------
__global__ __launch_bounds__(256) void gat_alpha(
    const float* __restrict__ xp, const float* __restrict__ a_src,
    const float* __restrict__ a_dst, float* __restrict__ as,
    float* __restrict__ ad, int n)
{
    int i = blockIdx.x * blockDim.x + threadIdx.x;
    if (i >= n * 4) return;
    int node = i >> 2, h = i & 3;
    const float* xr = xp + (size_t)node * 128 + h * 32;
    const float* s  = a_src + h * 32;
    const float* d  = a_dst + h * 32;
    float ss = 0.0f, dd = 0.0f;
#pragma unroll
    for (int f = 0; f < 32; ++f) {
        float v = xr[f];
        ss += v * s[f];
        dd += v * d[f];
    }
    as[i] = ss;
    ad[i] = dd;
}

// ---------------------------------------------------------------------------
// Kernel 3: init agg=0, maxb=-inf, denom=0
// ---------------------------------------------------------------------------
__global__ __launch_bounds__(256) void gat_init(
    float* __restrict__ agg, float* __restrict__ maxb,
    float* __restrict__ denom, int n)
{
    int i = blockIdx.x * blockDim.x + threadIdx.x;
    if (i < n * 128) agg[i] = 0.0f;
    if (i < n * 4) {
        maxb[i]  = __int_as_float(0xFF800000);  // -inf
        denom[i] = 0.0f;
    }
}

// ---------------------------------------------------------------------------
// Kernel 4: per-(edge,head) leaky-relu logits + segment max (atomic)
// Edges include N trailing self-loops.
// ---------------------------------------------------------------------------
__global__ __launch_bounds__(256) void gat_logit_max(
    const long long* __restrict__ ei, const float* __restrict__ as,
    const float* __restrict__ ad, float* __restrict__ exl,
    float* __restrict__ maxb, int E, int n)
{
    int i = blockIdx.x * blockDim.x + threadIdx.x;
    if (i >= (E + n) * 4) return;
    int e = i >> 2, h = i & 3;
    long long s, d;
    if (e < E) { s = ei[e]; d = ei[E + e]; } else { s = d = (long long)(e - E); }
    float l = as[s * 4 + h] + ad[d * 4 + h];
    l = (l > 0.0f) ? l : NEG_SLOPE * l;
    exl[i] = l;
    atomicMaxF(&maxb[d * 4 + h], l);
}

// ---------------------------------------------------------------------------
// Kernel 5: ex = exp(logit - max[dst]); denom[dst] += ex (atomic)
// ---------------------------------------------------------------------------
__global__ __launch_bounds__(256) void gat_exp_sum(
    const long long* __restrict__ ei, float* __restrict__ exl,
    const float* __restrict__ maxb, float* __restrict__ denom, int E, int n)
{
    int i = blockIdx.x * blockDim.x + threadIdx.x;
    if (i >= (E + n) * 4) return;
    int e = i >> 2, h = i & 3;
    long long d = (e < E) ? ei[E + e] : (long long)(e - E);
    float ex = expf(exl[i] - maxb[d * 4 + h]);
    exl[i] = ex;
    atomicAdd(&denom[d * 4 + h], ex);
}

// ---------------------------------------------------------------------------
// Kernel 6: one wave32 per edge. agg[dst] += coef * xp[src] (128 f32 atomics).
// xp/agg (51 MB each) stay L2-resident on the 192 MB L2.
// ---------------------------------------------------------------------------
__global__ __launch_bounds__(256) void gat_aggregate(
    const long long* __restrict__ ei, const float* __restrict__ xp,
    const float* __restrict__ exl, const float* __restrict__ denom,
    float* __restrict__ agg, int E, int n)
{
    int gt   = blockIdx.x * blockDim.x + threadIdx.x;
    int e    = gt >> 5;
    int lane = gt & 31;
    if (e >= E + n) return;
    long long s, d;
    if (e < E) { s = ei[e]; d = ei[E + e]; } else { s = d = (long long)(e - E); }
    const float* xs = xp + (size_t)s * 128;
    float* ag = agg + (size_t)d * 128;
#pragma unroll
    for (int h = 0; h < 4; ++h) {
        float coef = exl[e * 4 + h] / (denom[d * 4 + h] + 1e-16f);
        atomicAdd(&ag[h * 32 + lane], coef * xs[h * 32 + lane]);
    }
}

// ---------------------------------------------------------------------------
// Kernel 7: out = LN( (agg+bias) @ proj_w + proj_b + x ) * g + b
// Same WMMA structure as kernel 1; epilogue fused fully in registers with
// wave32 half-wave shuffle reductions (C-fragment rows live in 16 lanes).
// ---------------------------------------------------------------------------
__global__ __launch_bounds__(256) void gat_proj_ln(
    const float* __restrict__ agg, const float* __restrict__ bias,
    const float* __restrict__ pw, const float* __restrict__ pb,
    const float* __restrict__ x, const float* __restrict__ g,
    const float* __restrict__ bln, float* __restrict__ out, int nrows)
{
    __shared__ float Pt[128 * 128];   // Pt[col][d] = proj_w[d][col], swizzled
    int tid = threadIdx.x;
    for (int i = tid; i < 128 * 128; i += 256) {
        int c = i >> 7, d = i & 127;
        Pt[swz(c, d)] = pw[d * 128 + c];
    }
    __syncthreads();

    int lane  = tid & 31;
    int wave  = tid >> 5;
    int rb    = (blockIdx.x * 8 + wave) * 16;
    if (rb >= nrows) return;

    int m     = lane & 15;
    int khalf = lane >> 4;

    v8f acc[8];
#pragma unroll
    for (int t = 0; t < 8; ++t) acc[t] = (v8f)0.0f;

    int rload = rb + m; if (rload >= nrows) rload = nrows - 1;
    const float* ar = agg  + (size_t)rload * 128 + 2 * khalf;
    const float* br = bias + 2 * khalf;

    for (int k = 0; k < 128; k += 4) {
        v2f a;
        a.x = ar[k]     + br[k];
        a.y = ar[k + 1] + br[k + 1];
        int d0 = k + 2 * khalf;
#pragma unroll
        for (int t = 0; t < 8; ++t) {
            int col = t * 16 + m;
            v2f b = *(const v2f*)&Pt[swz(col, d0)];   // one ds_load_b64
            acc[t] = __builtin_amdgcn_wmma_f32_16x16x4_f32(
                false, a, false, b, (short)0, acc[t], false, false);
        }
    }

    // fused bias + residual + LayerNorm
    int roff = rb + khalf * 8;
#pragma unroll
    for (int v = 0; v < 8; ++v) {
        int r  = roff + v;
        int rl = (r < nrows) ? r : (nrows - 1);
        float rowvals[8];
        float sum = 0.0f, sq = 0.0f;
#pragma unroll
        for (int t = 0; t < 8; ++t) {
            int col   = t * 16 + m;
            float val = acc[t][v] + pb[col] + x[(size_t)rl * 128 + col];
            rowvals[t] = val;
            sum += val;
            sq  += val * val;
        }
        // reduce across the 16-lane half that owns this row (wave32)
#pragma unroll
        for (int off = 1; off < 16; off <<= 1) {
            sum += __shfl_xor(sum, off, 32);
            sq  += __shfl_xor(sq,  off, 32);
        }
        float mu   = sum * (1.0f / 128.0f);
        float var  = sq * (1.0f / 128.0f) - mu * mu;
        float rstd = rsqrtf(var + LN_EPS);
        if (r < nrows) {
#pragma unroll
            for (int t = 0; t < 8; ++t) {
                int col = t * 16 + m;
                out[(size_t)r * 128 + col] =
                    (rowvals[t] - mu) * rstd * g[col] + bln[col];
            }
        }
    }
}

// ---------------------------------------------------------------------------
extern "C" void kernel_launch(void* const* d_in, const int* in_sizes, int n_in,
                              void* d_out, int out_size, void* d_ws, size_t ws_size,
                              hipStream_t stream)
{
    const float*     x      = (const float*)d_in[0];
    const long long* ei     = (const long long*)d_in[1];
    const float*     W      = (const float*)d_in[2];
    const float*     a_src  = (const float*)d_in[3];
    const float*     a_dst  = (const float*)d_in[4];
    const float*     bias   = (const float*)d_in[5];
    const float*     proj_w = (const float*)d_in[6];
    const float*     proj_b = (const float*)d_in[7];
    const float*     ln_g   = (const float*)d_in[8];
    const float*     ln_b   = (const float*)d_in[9];
    float* out = (float*)d_out;

    const int N    = in_sizes[0] / 128;
    const int E    = in_sizes[1] / 2;
    const int Etot = E + N;

    // carve workspace (256B aligned)
    char* ws = (char*)d_ws;
    size_t off = 0;
    auto carve = [&](size_t bytes) -> float* {
        float* p = (float*)(ws + off);
        off += (bytes + 255) & ~(size_t)255;
        return p;
    };
    float* xp    = carve((size_t)N * 128 * sizeof(float));
    float* agg   = carve((size_t)N * 128 * sizeof(float));
    float* as    = carve((size_t)N * 4 * sizeof(float));
    float* ad    = carve((size_t)N * 4 * sizeof(float));
    float* maxb  = carve((size_t)N * 4 * sizeof(float));
    float* denom = carve((size_t)N * 4 * sizeof(float));
    float* exl   = carve((size_t)Etot * 4 * sizeof(float));
    (void)ws_size; (void)n_in; (void)out_size;

    const int rowBlocks = (N + 127) / 128;      // 128 rows per 256-thread block

    gat_gemm_xp<<<rowBlocks, 256, 0, stream>>>(x, W, xp, N);

    gat_alpha<<<(N * 4 + 255) / 256, 256, 0, stream>>>(xp, a_src, a_dst, as, ad, N);

    gat_init<<<((unsigned)(((size_t)N * 128 + 255) / 256)), 256, 0, stream>>>(agg, maxb, denom, N);

    gat_logit_max<<<(Etot * 4 + 255) / 256, 256, 0, stream>>>(ei, as, ad, exl, maxb, E, N);

    gat_exp_sum<<<(Etot * 4 + 255) / 256, 256, 0, stream>>>(ei, exl, maxb, denom, E, N);

    {
        long long threads = (long long)Etot * 32;
        unsigned blocks = (unsigned)((threads + 255) / 256);
        gat_aggregate<<<blocks, 256, 0, stream>>>(ei, xp, exl, denom, agg, E, N);
    }

    gat_proj_ln<<<rowBlocks, 256, 0, stream>>>(agg, bias, proj_w, proj_b, x, ln_g, ln_b, out, N);
}